// Sparse3DNA_34248069218387
// MI455X (gfx1250) — compile-verified
//
#include <hip/hip_runtime.h>

// ---------------------------------------------------------------------------
// Sparse 3D neighborhood attention (Sparse3DNA) for gfx1250 / MI455X.
//   HEADS=8, DIM_HEAD=64, KS=3, VIDEO=(8,32,32), DIM=512
//   n = 8193 tokens (BOS + 8192), kv padded to 9217 tokens (9 frames).
// Pipeline:
//   1) q  = (x[1:] @ Wq) * scale            -> fp32 WMMA GEMM  [8192 x 512]
//   2) kv = pad(x) @ Wkv                    -> fp32 WMMA GEMM  [9217 x 1024]
//   3) ao[0] = v_bos                        -> tiny copy
//   4) ao[1+qi] = softmax(q.kN) @ vN        -> wave-per-(query,head) kernel
//   5) out = ao @ Wo + bo                   -> fp32 WMMA GEMM  [8193 x 512]
// GEMM staging uses GLOBAL_LOAD_ASYNC_TO_LDS_B128 (ASYNCcnt) where the tile
// is fully in-bounds, with a guarded VGPR fallback on boundary tiles.
// ---------------------------------------------------------------------------

typedef __attribute__((ext_vector_type(2))) float v2f;
typedef __attribute__((ext_vector_type(8))) float v8f;

#define N_TOK   8193
#define NQ      8192
#define NKV     9217
#define DIM_    512
#define KVDIM   1024
#define NHEADS  8
#define DHEAD   64

// GEMM tiling
#define BM      128       // block tile M (8 waves x 16)
#define BN      64        // block tile N (4 wmma tiles per wave)
#define KC      32        // K chunk staged in LDS
#define APANEL  (BM * KC) // 4096 floats = 16 KB, row-major [BM][KC]
#define BPANEL  (KC * BN) // 2048 floats =  8 KB, row-major [KC][BN]

__device__ __forceinline__ void async_ld_b128(unsigned lds_off, const void* g)
{
    asm volatile("global_load_async_to_lds_b128 %0, %1, off"
                 :: "v"(lds_off), "v"(g) : "memory");
}
__device__ __forceinline__ void wait_asynccnt0()
{
    asm volatile("s_wait_asynccnt 0x0" ::: "memory");
}

// ---------------------------------------------------------------------------
// fp32 WMMA GEMM:  C[M,N] = alpha * A[M,K] @ B[K,N] (+ bias[N])
// A rows >= Mvalid are treated as zero (handles the kv zero-padding).
// Double-buffered LDS staging, async global->LDS copies on interior tiles.
//
// Fragment layouts per CDNA5 ISA 7.12.2 (wave32):
//   A 16x4 : lane%16 = M row; lane/16 selects K pair {2h, 2h+1}
//            row-major LDS A => one aligned ds_load_b64 per fragment
//   B 4x16 : lane%16 = N col; lane/16 selects K pair {2h, 2h+1}
//            row-major LDS B => two b32 reads, BN*4 bytes apart
//   C 16x16: lane%16 = N col; VGPR r holds M = r + 8*(lane/16)
// ---------------------------------------------------------------------------
__global__ __launch_bounds__(256) void wmma_gemm_f32(
    const float* __restrict__ A, int Mvalid,
    const float* __restrict__ B,
    float* __restrict__ C,
    const float* __restrict__ bias,
    int M, int N, int K, float alpha)
{
    __shared__ __align__(16) float lds[2][APANEL + BPANEL];

    const int tid  = threadIdx.x;
    const int lane = tid & 31;
    const int wave = tid >> 5;                 // 0..7
    const int m0b  = blockIdx.x * BM;
    const int n0   = blockIdx.y * BN;
    const int lm   = lane & 15;                // row (A) / col (B,C)
    const int hi   = lane >> 4;                // K-pair selector

    // Interior tile: every A row of this block is a real (non-pad) row.
    const bool a_all_valid = (m0b + BM) <= Mvalid;

    v8f acc[4] = {};

    // ---- cooperative fill of one LDS buffer with the [k0, k0+KC) panels ----
    auto fill = [&](float* buf, int k0) {
        float* ldsA = buf;
        float* ldsB = buf + APANEL;
        // ---- B panel: KC x BN, always in-bounds -> async copy (16B/lane) --
        {
            const unsigned b_base = (unsigned)(uintptr_t)ldsB;
#pragma unroll
            for (int i = 0; i < 2; ++i) {
                const int chunk = (wave * 2 + i) * 32 + lane;   // 0..511
                const int kr = chunk >> 4;                      // 0..31
                const int cq = (chunk & 15) << 2;               // 0,4,..,60
                async_ld_b128(b_base + (unsigned)chunk * 16,
                              B + (size_t)(k0 + kr) * N + n0 + cq);
            }
        }
        // ---- A panel: BM x KC ---------------------------------------------
        if (a_all_valid) {
#pragma unroll
            for (int i = 0; i < 4; ++i) {
                const int chunk = (wave * 4 + i) * 32 + lane;   // 0..1023
                const int row = chunk >> 3;                     // 0..127
                const int kq  = (chunk & 7) << 2;               // 0,4,..,28
                async_ld_b128((unsigned)(uintptr_t)ldsA + (unsigned)chunk * 16,
                              A + (size_t)(m0b + row) * K + k0 + kq);
            }
        } else {
            // guarded fallback: pad rows -> zeros (via VGPRs)
#pragma unroll
            for (int i = 0; i < 4; ++i) {
                const int idx4 = tid + i * 256;                 // 0..1023
                const int row  = idx4 >> 3;
                const int kq   = (idx4 & 7) << 2;
                const int grow = m0b + row;
                const bool ok  = (grow < Mvalid);
                const float4 v = *reinterpret_cast<const float4*>(
                    A + (size_t)(ok ? grow : 0) * K + k0 + kq);
                float4 w;
                w.x = ok ? v.x : 0.0f;  w.y = ok ? v.y : 0.0f;
                w.z = ok ? v.z : 0.0f;  w.w = ok ? v.w : 0.0f;
                *reinterpret_cast<float4*>(&ldsA[row * KC + kq]) = w;
            }
        }
    };

    fill(lds[0], 0);
    wait_asynccnt0();
    __syncthreads();

    for (int kc = 0; kc < K; kc += KC) {
        const int cur = (kc / KC) & 1;
        if (kc + KC < K)
            fill(lds[cur ^ 1], kc + KC);           // prefetch next chunk

        const float* ldsA = lds[cur];
        const float* ldsB = lds[cur] + APANEL;
#pragma unroll
        for (int kk = 0; kk < KC; kk += 4) {
            const int ke = kk + 2 * hi;            // this lane's K pair base
            const v2f a = *reinterpret_cast<const v2f*>(
                &ldsA[(wave * 16 + lm) * KC + ke]);
#pragma unroll
            for (int t = 0; t < 4; ++t) {
                v2f b;
                b.x = ldsB[ke * BN + t * 16 + lm];
                b.y = ldsB[ke * BN + BN + t * 16 + lm];
                acc[t] = __builtin_amdgcn_wmma_f32_16x16x4_f32(
                    false, a, false, b, (short)0, acc[t], false, false);
            }
        }
        wait_asynccnt0();   // next-chunk async writes landed in LDS
        __syncthreads();    // cur fully read + next fully written
    }

    // ---- epilogue ----
    const int m0 = m0b + wave * 16;
#pragma unroll
    for (int t = 0; t < 4; ++t) {
        const int n = n0 + t * 16 + lm;
        const float bv = bias ? bias[n] : 0.0f;
#pragma unroll
        for (int r = 0; r < 8; ++r) {
            const int m = m0 + r + 8 * hi;
            if (m < M)
                C[(size_t)m * N + n] = alpha * acc[t][r] + bv;
        }
    }
}

// ---------------------------------------------------------------------------
// attn: one wave per (query, head). Lane 0 = BOS key, lanes 1..27 = the 27
// neighborhood keys (ordering j = a*9 + e*3 + g, matching unfold3d).
// ---------------------------------------------------------------------------
__global__ __launch_bounds__(256) void attn_kernel(
    const float* __restrict__ q,   // [NQ, 512]   (pre-scaled)
    const float* __restrict__ kv,  // [NKV, 1024] (k: 0..511, v: 512..1023)
    float* __restrict__ ao)        // [N_TOK, 512]
{
    const int qi   = blockIdx.x;          // 0..8191 (video token index)
    const int h    = threadIdx.x >> 5;    // head = wave id (0..7)
    const int lane = threadIdx.x & 31;

    const int f  = qi >> 10;
    const int y  = (qi >> 5) & 31;
    const int xx = qi & 31;

    // Key row in kv buffer (kv row 0 = BOS; video token t -> row t+1).
    int  krow  = 0;
    bool valid = (lane == 0);             // BOS always attended
    if (lane >= 1 && lane <= 27) {
        const int j  = lane - 1;
        const int a  = j / 9;
        const int e  = (j / 3) % 3;
        const int g  = j % 3;
        const int kf = f + a - 1;
        const int kh = y + e - 1;
        const int kw = xx + g - 1;
        const bool inb = (kf >= 0) & (kf < 8) & (kh >= 0) & (kh < 32) &
                         (kw >= 0) & (kw < 32);
        const int nidx = (kf << 10) + (kh << 5) + kw;
        valid = inb && (nidx <= qi);      // causal: masked when qi < nidx
        krow  = valid ? nidx + 1 : 0;
    }

    // score = q . k  (64-dim dot per lane)
    const float* qrow = q  + (size_t)qi   * DIM_  + h * DHEAD;
    const float* kptr = kv + (size_t)krow * KVDIM + h * DHEAD;
    float s = -3.402823466e+38f;
    if (valid) {
        float acc = 0.0f;
#pragma unroll
        for (int d = 0; d < DHEAD; d += 2) {
            const float2 qv = *reinterpret_cast<const float2*>(qrow + d);
            const float2 kx = *reinterpret_cast<const float2*>(kptr + d);
            acc += qv.x * kx.x + qv.y * kx.y;
        }
        s = acc;
    }

    // wave32 softmax over lanes (invalid lanes contribute 0)
    float mx = s;
#pragma unroll
    for (int off = 16; off > 0; off >>= 1)
        mx = fmaxf(mx, __shfl_xor(mx, off, 32));
    float e = valid ? __expf(s - mx) : 0.0f;
    float sum = e;
#pragma unroll
    for (int off = 16; off > 0; off >>= 1)
        sum += __shfl_xor(sum, off, 32);
    const float p = e / sum;

    // out[d] = sum_j p_j * v[key_j][d]; each lane owns 2 dims (coalesced)
    const int d0 = lane * 2;
    float acc0 = 0.0f, acc1 = 0.0f;
    const float* vbase = kv + DIM_ + h * DHEAD + d0;   // v half of kv rows
#pragma unroll
    for (int j = 0; j < 28; ++j) {
        const float pj = __shfl(p, j, 32);
        const int   rj = __shfl(krow, j, 32);
        const float2 vv =
            *reinterpret_cast<const float2*>(vbase + (size_t)rj * KVDIM);
        acc0 += pj * vv.x;
        acc1 += pj * vv.y;
    }

    float* orow = ao + (size_t)(qi + 1) * DIM_ + h * DHEAD;
    orow[d0]     = acc0;
    orow[d0 + 1] = acc1;
}

// Row 0 of the pre-projection output is v_bos (per-head concat = kv[0][512:]).
__global__ void bos_copy_kernel(const float* __restrict__ kv,
                                float* __restrict__ ao)
{
    const int c = threadIdx.x;            // 512 threads
    ao[c] = kv[DIM_ + c];
}

// ---------------------------------------------------------------------------
extern "C" void kernel_launch(void* const* d_in, const int* in_sizes, int n_in,
                              void* d_out, int out_size, void* d_ws, size_t ws_size,
                              hipStream_t stream)
{
    (void)in_sizes; (void)n_in; (void)out_size; (void)ws_size;

    const float* x   = (const float*)d_in[0];   // [8193, 512]
    const float* Wq  = (const float*)d_in[1];   // [512, 512]
    const float* Wkv = (const float*)d_in[2];   // [512, 1024]
    const float* Wo  = (const float*)d_in[3];   // [512, 512]
    const float* bo  = (const float*)d_in[4];   // [512]
    float* out = (float*)d_out;                 // [8193, 512]

    float* qb  = (float*)d_ws;                       // [8192, 512]
    float* kvb = qb  + (size_t)NQ  * DIM_;           // [9217, 1024]
    float* aob = kvb + (size_t)NKV * KVDIM;          // [8193, 512]

    const dim3 blk(256);
    const float scale = 0.125f;                      // 64^-0.5

    // q = x[1:8193] @ Wq * scale      (M=8192 -> 64 M-blocks, N=512 -> 8)
    wmma_gemm_f32<<<dim3(64, 8), blk, 0, stream>>>(
        x + DIM_, NQ, Wq, qb, nullptr, NQ, DIM_, DIM_, scale);

    // kv = pad(x) @ Wkv               (M=9217 -> 73 M-blocks, N=1024 -> 16)
    wmma_gemm_f32<<<dim3(73, 16), blk, 0, stream>>>(
        x, N_TOK, Wkv, kvb, nullptr, NKV, KVDIM, DIM_, 1.0f);

    // ao[0] = v_bos
    bos_copy_kernel<<<1, DIM_, 0, stream>>>(kvb, aob);

    // ao[1+qi] = attention(qi)        (one block per query, wave per head)
    attn_kernel<<<NQ, 256, 0, stream>>>(qb, kvb, aob);

    // out = ao @ Wo + bo              (M=8193 -> 65 M-blocks, N=512 -> 8)
    wmma_gemm_f32<<<dim3(65, 8), blk, 0, stream>>>(
        aob, N_TOK, Wo, out, bo, N_TOK, DIM_, DIM_, 1.0f);
}